// IterativeDecimator_63350767616370
// MI455X (gfx1250) — compile-verified
//
#include <hip/hip_runtime.h>
#include <hip/hip_bf16.h>
#include <math.h>

// ---------------------------------------------------------------------------
// Problem constants (from the reference)
// ---------------------------------------------------------------------------
#define D_FEAT   64
#define HIDDEN   32
#define KCL      16      // clusters
#define NPG      1000    // nodes per graph
#define EPG      5000    // edges per graph

typedef float v2f __attribute__((ext_vector_type(2)));
typedef float v8f __attribute__((ext_vector_type(8)));

// D = A(16x4 f32) * B(4x16 f32) + C(16x16 f32)   -- full fp32 matrix pipe
__device__ __forceinline__ v8f wmma_f32_k4(v2f a, v2f b, v8f c) {
    return __builtin_amdgcn_wmma_f32_16x16x4_f32(
        /*neg_a=*/false, a, /*neg_b=*/false, b,
        /*c_mod=*/(short)0, c, /*reuse_a=*/false, /*reuse_b=*/false);
}

// ---------------------------------------------------------------------------
// Kernel A: assignments = softmax(relu(X@W1+b1)@W2+b2)
// 256 threads = 8 waves; each wave owns a 16-node tile (128 nodes / block).
// Layer 1: two 16x16 WMMA accum tiles (hidden 0-15 / 16-31), K=64 (16 steps).
// Layer 2 re-stages h through LDS (D-layout -> A-layout transpose), K=32.
// Softmax over clusters = 16-lane shuffle reduction within each half-wave.
// ---------------------------------------------------------------------------
__global__ void __launch_bounds__(256)
assign_mlp_kernel(const float* __restrict__ nodes,
                  const float* __restrict__ w1, const float* __restrict__ b1,
                  const float* __restrict__ w2, const float* __restrict__ b2,
                  float* __restrict__ assign, int nTotal)
{
    __shared__ float hbuf[8][16][HIDDEN + 1];   // +1 pad vs bank conflicts

    const int tid  = threadIdx.x;
    const int wave = tid >> 5;
    const int lane = tid & 31;
    const int half = lane >> 4;      // which 16-lane half
    const int lx   = lane & 15;
    const int nodeBase = blockIdx.x * 128 + wave * 16;

    // A operand row for this lane (M = lx); clamp OOB tail (reads only)
    int rowA = nodeBase + lx;
    if (rowA > nTotal - 1) rowA = nTotal - 1;

    // ---- layer 1: [16 x 64] @ [64 x 32] -------------------------------
    v8f c0 = {}; v8f c1 = {};
    #pragma unroll
    for (int kk = 0; kk < 16; ++kk) {
        const int k0 = 4 * kk + 2 * half;              // K = k0 + j
        v2f a;  a.x  = nodes[rowA * D_FEAT + k0];
                a.y  = nodes[rowA * D_FEAT + k0 + 1];
        v2f b0; b0.x = w1[(k0    ) * HIDDEN + lx];
                b0.y = w1[(k0 + 1) * HIDDEN + lx];
        v2f bt; bt.x = w1[(k0    ) * HIDDEN + 16 + lx];
                bt.y = w1[(k0 + 1) * HIDDEN + 16 + lx];
        c0 = wmma_f32_k4(a, b0, c0);
        c1 = wmma_f32_k4(a, bt, c1);
    }

    // bias + relu, stage h into LDS as [node][hidden] for the transpose
    #pragma unroll
    for (int r = 0; r < 8; ++r) {
        const int m = r + 8 * half;                    // node row in tile
        float v0 = c0[r] + b1[lx];        v0 = v0 > 0.f ? v0 : 0.f;
        float v1 = c1[r] + b1[16 + lx];   v1 = v1 > 0.f ? v1 : 0.f;
        hbuf[wave][m][lx]      = v0;
        hbuf[wave][m][16 + lx] = v1;
    }
    __syncthreads();

    // ---- layer 2: [16 x 32] @ [32 x 16] -------------------------------
    v8f c2 = {};
    #pragma unroll
    for (int kk = 0; kk < 8; ++kk) {
        const int k0 = 4 * kk + 2 * half;
        v2f a;  a.x = hbuf[wave][lx][k0];
                a.y = hbuf[wave][lx][k0 + 1];
        v2f b;  b.x = w2[(k0    ) * KCL + lx];
                b.y = w2[(k0 + 1) * KCL + lx];
        c2 = wmma_f32_k4(a, b, c2);
    }

    // ---- softmax over clusters (N dim = 16 lanes of this half) --------
    const float bias2 = b2[lx];
    #pragma unroll
    for (int r = 0; r < 8; ++r) {
        float v  = c2[r] + bias2;
        float mx = v;
        mx = fmaxf(mx, __shfl_xor(mx, 1, 16));
        mx = fmaxf(mx, __shfl_xor(mx, 2, 16));
        mx = fmaxf(mx, __shfl_xor(mx, 4, 16));
        mx = fmaxf(mx, __shfl_xor(mx, 8, 16));
        float e = __expf(v - mx);
        float s = e;
        s += __shfl_xor(s, 1, 16);
        s += __shfl_xor(s, 2, 16);
        s += __shfl_xor(s, 4, 16);
        s += __shfl_xor(s, 8, 16);
        const float aout = e / s;
        const int node = nodeBase + r + 8 * half;
        if (node < nTotal) assign[node * KCL + lx] = aout;
    }
}

// ---------------------------------------------------------------------------
// Kernel B: coarse_nodes[g] = A_g^T @ X_g    ([16 x 1000] @ [1000 x 64])
// One block (8 waves) per graph; K-steps (4 nodes each) strided across waves;
// 4 N-tiles (feat 0-15..48-63) per wave; serial LDS sum across waves.
// ---------------------------------------------------------------------------
__global__ void __launch_bounds__(256)
coarse_nodes_kernel(const float* __restrict__ nodes,
                    const float* __restrict__ assign,
                    float* __restrict__ out /* [G*16*64] */)
{
    __shared__ float red[8][KCL][D_FEAT];   // 32 KB

    const int g    = blockIdx.x;
    const int tid  = threadIdx.x;
    const int wave = tid >> 5;
    const int lane = tid & 31;
    const int half = lane >> 4;
    const int lx   = lane & 15;

    v8f acc0 = {}, acc1 = {}, acc2 = {}, acc3 = {};
    const int steps = NPG / 4;                         // 250
    for (int s = wave; s < steps; s += 8) {
        const int nb = g * NPG + 4 * s + 2 * half;     // node idx for j=0
        // A = assignments^T: A[m=cluster, k=node] -> 64B coalesced row reads
        v2f a; a.x = assign[(nb    ) * KCL + lx];
               a.y = assign[(nb + 1) * KCL + lx];
        v2f b;
        b.x = nodes[(nb) * D_FEAT +  0 + lx]; b.y = nodes[(nb + 1) * D_FEAT +  0 + lx];
        acc0 = wmma_f32_k4(a, b, acc0);
        b.x = nodes[(nb) * D_FEAT + 16 + lx]; b.y = nodes[(nb + 1) * D_FEAT + 16 + lx];
        acc1 = wmma_f32_k4(a, b, acc1);
        b.x = nodes[(nb) * D_FEAT + 32 + lx]; b.y = nodes[(nb + 1) * D_FEAT + 32 + lx];
        acc2 = wmma_f32_k4(a, b, acc2);
        b.x = nodes[(nb) * D_FEAT + 48 + lx]; b.y = nodes[(nb + 1) * D_FEAT + 48 + lx];
        acc3 = wmma_f32_k4(a, b, acc3);
    }

    #pragma unroll
    for (int r = 0; r < 8; ++r) {
        const int m = r + 8 * half;                    // cluster row
        red[wave][m][ 0 + lx] = acc0[r];
        red[wave][m][16 + lx] = acc1[r];
        red[wave][m][32 + lx] = acc2[r];
        red[wave][m][48 + lx] = acc3[r];
    }
    __syncthreads();

    for (int o = tid; o < KCL * D_FEAT; o += 256) {
        const int m = o >> 6, d = o & 63;
        float s = 0.f;
        #pragma unroll
        for (int w = 0; w < 8; ++w) s += red[w][m][d];
        out[(g * KCL + m) * D_FEAT + d] = s;
    }
}

// ---------------------------------------------------------------------------
// Kernel C: coarse_adj[g] = S_s^T @ S_r  ([16 x 5000] @ [5000 x 16], gathered)
// plus the trivial c_senders / c_receivers / c_n_node / c_n_edge outputs.
// ---------------------------------------------------------------------------
__global__ void __launch_bounds__(256)
coarse_adj_kernel(const float* __restrict__ assign,
                  const int* __restrict__ senders,
                  const int* __restrict__ receivers,
                  float* __restrict__ edge_w,   // [G*256]
                  float* __restrict__ cs,       // [G*256]
                  float* __restrict__ cr,       // [G*256]
                  float* __restrict__ cnn,      // [G]
                  float* __restrict__ cne)      // [G]
{
    __shared__ float red[8][KCL][KCL];   // 8 KB

    const int g    = blockIdx.x;
    const int tid  = threadIdx.x;
    const int wave = tid >> 5;
    const int lane = tid & 31;
    const int half = lane >> 4;
    const int lx   = lane & 15;

    v8f acc = {};
    const int steps = EPG / 4;                         // 1250
    for (int t = wave; t < steps; t += 8) {
        const int eb = g * EPG + 4 * t + 2 * half;     // edge idx for j=0
        const int s0 = senders[eb],     s1 = senders[eb + 1];
        const int r0 = receivers[eb],   r1 = receivers[eb + 1];
        v2f a; a.x = assign[s0 * KCL + lx];  a.y = assign[s1 * KCL + lx];
        v2f b; b.x = assign[r0 * KCL + lx];  b.y = assign[r1 * KCL + lx];
        acc = wmma_f32_k4(a, b, acc);
    }

    #pragma unroll
    for (int r = 0; r < 8; ++r)
        red[wave][r + 8 * half][lx] = acc[r];
    __syncthreads();

    const int s = tid >> 4, r = tid & 15;              // tid == s*16 + r
    float sum = 0.f;
    #pragma unroll
    for (int w = 0; w < 8; ++w) sum += red[w][s][r];
    edge_w[g * 256 + tid] = sum;
    cs[g * 256 + tid] = (float)(g * KCL + s);
    cr[g * 256 + tid] = (float)(g * KCL + r);
    if (tid == 0) { cnn[g] = (float)KCL; cne[g] = (float)(KCL * KCL); }
}

// ---------------------------------------------------------------------------
// Host launcher
// ---------------------------------------------------------------------------
extern "C" void kernel_launch(void* const* d_in, const int* in_sizes, int n_in,
                              void* d_out, int out_size, void* d_ws, size_t ws_size,
                              hipStream_t stream)
{
    const float* nodes     = (const float*)d_in[0];
    const float* w1        = (const float*)d_in[1];
    const float* b1        = (const float*)d_in[2];
    const float* w2        = (const float*)d_in[3];
    const float* b2        = (const float*)d_in[4];
    const int*   senders   = (const int*)d_in[5];
    const int*   receivers = (const int*)d_in[6];

    const int N = in_sizes[0] / D_FEAT;   // 100000
    const int G = in_sizes[7];            // 100

    float* out = (float*)d_out;
    // flat output layout in reference return order
    const long off_cn   = 0;                                   // coarse_nodes  G*16*64
    const long off_ew   = off_cn + (long)G * KCL * D_FEAT;     // edge weights  G*256
    const long off_as   = off_ew + (long)G * KCL * KCL;        // assignments   N*16
    const long off_cs   = off_as + (long)N * KCL;              // c_senders     G*256
    const long off_cr   = off_cs + (long)G * KCL * KCL;        // c_receivers   G*256
    const long off_cnn  = off_cr + (long)G * KCL * KCL;        // c_n_node      G
    const long off_cne  = off_cnn + (long)G;                   // c_n_edge      G

    float* assign = out + off_as;

    const int blocksA = (N + 127) / 128;
    assign_mlp_kernel<<<blocksA, 256, 0, stream>>>(nodes, w1, b1, w2, b2, assign, N);

    coarse_nodes_kernel<<<G, 256, 0, stream>>>(nodes, assign, out + off_cn);

    coarse_adj_kernel<<<G, 256, 0, stream>>>(assign, senders, receivers,
                                             out + off_ew, out + off_cs, out + off_cr,
                                             out + off_cnn, out + off_cne);
}